// Attention_29832842838823
// MI455X (gfx1250) — compile-verified
//
#include <hip/hip_runtime.h>
#include <hip/hip_bf16.h>
#include <math.h>

typedef __attribute__((ext_vector_type(2))) float v2f;
typedef __attribute__((ext_vector_type(4))) float v4f;
typedef __attribute__((ext_vector_type(8))) float v8f;

#define N_EMBD 2048
#define N_HEAD 32
#define SEQ_T 2048
#define HEAD_D 64
#define BATCH 4
#define THREE_C (3 * N_EMBD)

// D = A(16x4) * B(4x16) + C, full fp32 (V_WMMA_F32_16X16X4_F32)
__device__ __forceinline__ v8f wmma4(v2f a, v2f b, v8f c) {
  return __builtin_amdgcn_wmma_f32_16x16x4_f32(false, a, false, b, (short)0, c,
                                               false, false);
}

// CDNA5 async DMA: global -> LDS, 16B per lane, tracked by ASYNCcnt.
__device__ __forceinline__ uint32_t lds_off(const void* p) {
  // generic LDS pointer = SHARED aperture | 32-bit LDS offset (ISA 10.2)
  return (uint32_t)(uintptr_t)p;
}
__device__ __forceinline__ void async_b128(uint32_t lds, const float* g) {
  asm volatile("global_load_async_to_lds_b128 %0, %1, off" ::"v"(lds), "v"(g)
               : "memory");
}
__device__ __forceinline__ void wait_async0() {
  asm volatile("s_wait_asynccnt 0x0" ::: "memory");
}

// ---------------- RoPE cache: rope[t*64 + 2i] = cos, [.. + 1] = sin ---------
__global__ void rope_cache_kernel(float* __restrict__ rope) {
  int idx = blockIdx.x * blockDim.x + threadIdx.x;  // 0 .. T*32-1
  if (idx >= SEQ_T * 32) return;
  int t = idx >> 5;
  int i = idx & 31;
  float freq = __powf(10000.0f, -(2.0f * (float)i) / 64.0f);
  float ang = (float)t * freq;
  float s, c;
  __sincosf(ang, &s, &c);
  rope[idx * 2 + 0] = c;
  rope[idx * 2 + 1] = s;
}

// ---------------- Tiled GEMM: C[M,N] = A[M,K] * B[K,N] + bias --------------
// 256 threads (8 waves), tile 128x128, KC=32, double-buffered async DMA
// staging. Wave = 64x32 region = 4x2 grid of 16x16 WMMA-f32 tiles, K-step 4.
__global__ __launch_bounds__(256) void gemm_bias_kernel(
    const float* __restrict__ A, const float* __restrict__ B,
    const float* __restrict__ bias, float* __restrict__ C, int M, int N,
    int K) {
  constexpr int KC = 32;
  constexpr int LDA = KC + 4;   // 36 floats (144B rows, 16B aligned)
  constexpr int LDB = 128 + 4;  // 132 floats (528B rows, 16B aligned)
  __shared__ float Asm[2][128 * LDA];
  __shared__ float Bsm[2][KC * LDB];

  const int tid = threadIdx.x;
  const int m0 = blockIdx.y * 128;
  const int n0 = blockIdx.x * 128;
  const int wave = tid >> 5, lane = tid & 31;
  const int lm = lane & 15, half = lane >> 4;
  const int wm = wave >> 2, wn = wave & 3;  // 2 x 4 wave grid

  // per-thread DMA addressing: A tile 128x32 (4 x b128), B tile 32x128 (4x)
  const int arow = tid >> 3, acol = (tid & 7) * 4;
  const int brow = tid >> 5, bcol = (tid & 31) * 4;

  v8f acc[4][2] = {};
  const int nk = K / KC;

  // prologue: async-stage chunk 0 into buffer 0
#pragma unroll
  for (int r = 0; r < 4; ++r)
    async_b128(lds_off(&Asm[0][(arow + r * 32) * LDA + acol]),
               &A[(size_t)(m0 + arow + r * 32) * K + acol]);
#pragma unroll
  for (int r = 0; r < 4; ++r)
    async_b128(lds_off(&Bsm[0][(brow + r * 8) * LDB + bcol]),
               &B[(size_t)(brow + r * 8) * N + n0 + bcol]);

  for (int kt = 0; kt < nk; ++kt) {
    const int buf = kt & 1;
    wait_async0();    // this wave's tile-kt DMA complete
    __syncthreads();  // all waves' DMA visible; all done reading buf^1
    if (kt + 1 < nk) {
      const int k1 = (kt + 1) * KC;  // overlap next-tile DMA with compute
#pragma unroll
      for (int r = 0; r < 4; ++r)
        async_b128(lds_off(&Asm[buf ^ 1][(arow + r * 32) * LDA + acol]),
                   &A[(size_t)(m0 + arow + r * 32) * K + k1 + acol]);
#pragma unroll
      for (int r = 0; r < 4; ++r)
        async_b128(lds_off(&Bsm[buf ^ 1][(brow + r * 8) * LDB + bcol]),
                   &B[(size_t)(k1 + brow + r * 8) * N + n0 + bcol]);
    }
    const float* As = Asm[buf];
    const float* Bs = Bsm[buf];
#pragma unroll
    for (int kk = 0; kk < KC; kk += 4) {
      const int bk = kk + half * 2;  // A frag: lanes0-15 -> K 0,1; 16-31 -> 2,3
      v2f a[4], b[2];
#pragma unroll
      for (int i = 0; i < 4; ++i)
        a[i] = *(const v2f*)&As[(wm * 64 + i * 16 + lm) * LDA + bk];
#pragma unroll
      for (int j = 0; j < 2; ++j) {
        v2f t;
        t.x = Bs[bk * LDB + wn * 32 + j * 16 + lm];
        t.y = Bs[(bk + 1) * LDB + wn * 32 + j * 16 + lm];
        b[j] = t;
      }
#pragma unroll
      for (int i = 0; i < 4; ++i)
#pragma unroll
        for (int j = 0; j < 2; ++j) acc[i][j] = wmma4(a[i], b[j], acc[i][j]);
    }
  }

  // epilogue: C/D layout: VGPR r -> row (r + 8*half), col = lane%16
#pragma unroll
  for (int i = 0; i < 4; ++i)
#pragma unroll
    for (int j = 0; j < 2; ++j) {
      int col = n0 + wn * 32 + j * 16 + lm;
      float bv = bias[col];
#pragma unroll
      for (int r = 0; r < 8; ++r) {
        int row = m0 + wm * 64 + i * 16 + r + 8 * half;
        C[(size_t)row * N + col] = acc[i][j][r] + bv;
      }
    }
}

// ---------------- Flash attention, WG = (b,h,qblock of 64), 4 waves --------
__global__ __launch_bounds__(128) void flash_attn_kernel(
    const float* __restrict__ qkv, const float* __restrict__ rope,
    float* __restrict__ y) {
  constexpr int LDK = HEAD_D + 4;  // 68 floats (272B rows, 16B aligned)
  __shared__ float Ksm[64 * LDK];
  __shared__ float Vsm[64 * LDK];
  __shared__ float Psm[4 * 16 * LDK];  // per-wave C->A layout bounce

  const int tid = threadIdx.x;
  const int wave = tid >> 5, lane = tid & 31;
  const int lm = lane & 15, half = lane >> 4;
  const int bh = blockIdx.x;
  const int b = bh / N_HEAD, h = bh % N_HEAD;
  const int qb = blockIdx.y;
  const float scale = 0.125f;  // 1/sqrt(64)

  // Q fragments (16 K-steps of 4), RoPE + scale fused at load.
  // A-frag: lane holds {Q[m][d], Q[m][d+1]} with d even -> exact RoPE pair.
  const int mrow_g = qb * 64 + wave * 16 + lm;
  v2f qf[16];
  {
    const float* qrow =
        qkv + (size_t)(b * SEQ_T + mrow_g) * THREE_C + h * HEAD_D;
    const float* rrow = rope + (size_t)mrow_g * 64;
#pragma unroll
    for (int s = 0; s < 16; ++s) {
      int d = s * 4 + half * 2;
      v2f x = *(const v2f*)&qrow[d];
      float c = rrow[d], sn = rrow[d + 1];
      v2f r;
      r.x = (x.x * c - x.y * sn) * scale;
      r.y = (x.y * c + x.x * sn) * scale;
      qf[s] = r;
    }
  }

  float mrow[8], lrow[8];
  v8f oacc[4] = {};
#pragma unroll
  for (int r = 0; r < 8; ++r) {
    mrow[r] = -1e30f;
    lrow[r] = 0.0f;
  }
  const int wbase = wave * 16 * LDK;

  for (int kb = 0; kb <= qb; ++kb) {
    __syncthreads();  // all readers of Ksm/Vsm done before overwrite
    // V tile (64x64) via async DMA: 8 x b128 per thread
#pragma unroll
    for (int i = 0; i < 8; ++i) {
      int flat = i * 128 + tid;  // 1024 chunks of 4 floats
      int row = flat >> 4, c4 = (flat & 15) * 4;
      async_b128(lds_off(&Vsm[row * LDK + c4]),
                 &qkv[(size_t)(b * SEQ_T + kb * 64 + row) * THREE_C +
                      2 * N_EMBD + h * HEAD_D + c4]);
    }
    // K tile (64x64) with RoPE applied in flight
#pragma unroll
    for (int i = 0; i < 16; ++i) {
      int flat = i * 128 + tid;
      int key = flat >> 5;
      int p = flat & 31;  // dim pair index
      int t = kb * 64 + key;
      v2f kx = *(const v2f*)&qkv[(size_t)(b * SEQ_T + t) * THREE_C + N_EMBD +
                                 h * HEAD_D + 2 * p];
      float c = rope[t * 64 + 2 * p];
      float sn = rope[t * 64 + 2 * p + 1];
      v2f kr;
      kr.x = kx.x * c - kx.y * sn;
      kr.y = kx.y * c + kx.x * sn;
      *(v2f*)&Ksm[key * LDK + 2 * p] = kr;
    }
    wait_async0();    // V DMA done (this wave)
    __syncthreads();  // publish K stores + all waves' V DMA

    // S (16 x 64) = Q * K^T : B-frag b.x = K[key=n][d=bk]
    v8f sacc[4] = {};
#pragma unroll
    for (int s = 0; s < 16; ++s) {
      int bk = s * 4 + half * 2;
#pragma unroll
      for (int n = 0; n < 4; ++n) {
        v2f bb = *(const v2f*)&Ksm[(n * 16 + lm) * LDK + bk];
        sacc[n] = wmma4(qf[s], bb, sacc[n]);
      }
    }

    // causal mask only on the diagonal block
    if (kb == qb) {
#pragma unroll
      for (int n = 0; n < 4; ++n) {
        int col = kb * 64 + n * 16 + lm;
#pragma unroll
        for (int r = 0; r < 8; ++r) {
          int row = qb * 64 + wave * 16 + r + 8 * half;
          sacc[n][r] = (col > row) ? -1e30f : sacc[n][r];
        }
      }
    }

    // online softmax; each row lives in one 16-lane half -> width-16 shfl
#pragma unroll
    for (int r = 0; r < 8; ++r) {
      float mx = sacc[0][r];
#pragma unroll
      for (int n = 1; n < 4; ++n) mx = fmaxf(mx, sacc[n][r]);
#pragma unroll
      for (int off = 1; off < 16; off <<= 1)
        mx = fmaxf(mx, __shfl_xor(mx, off, 32));
      float mnew = fmaxf(mrow[r], mx);
      float corr = __expf(mrow[r] - mnew);
      mrow[r] = mnew;
      float rsum = 0.0f;
#pragma unroll
      for (int n = 0; n < 4; ++n) {
        float pv = __expf(sacc[n][r] - mnew);
        sacc[n][r] = pv;
        rsum += pv;
      }
#pragma unroll
      for (int off = 1; off < 16; off <<= 1) rsum += __shfl_xor(rsum, off, 32);
      lrow[r] = lrow[r] * corr + rsum;
#pragma unroll
      for (int n = 0; n < 4; ++n) oacc[n][r] = oacc[n][r] * corr;
    }

    // bounce P through per-wave LDS to convert C-layout -> A-layout
#pragma unroll
    for (int n = 0; n < 4; ++n)
#pragma unroll
      for (int r = 0; r < 8; ++r)
        Psm[wbase + (r + 8 * half) * LDK + n * 16 + lm] = sacc[n][r];
    asm volatile("s_wait_dscnt 0x0" ::: "memory");  // wave-local LDS RAW fence

    // O (16 x 64) += P (16 x 64keys) * V (64keys x 64)
#pragma unroll
    for (int s = 0; s < 16; ++s) {
      int bk = s * 4 + half * 2;
      v2f ap = *(const v2f*)&Psm[wbase + lm * LDK + bk];
#pragma unroll
      for (int n = 0; n < 4; ++n) {
        v2f bv;
        bv.x = Vsm[bk * LDK + n * 16 + lm];
        bv.y = Vsm[(bk + 1) * LDK + n * 16 + lm];
        oacc[n] = wmma4(ap, bv, oacc[n]);
      }
    }
  }

  // normalize and write y[B,T,C] (head-interleaved layout for proj GEMM)
#pragma unroll
  for (int r = 0; r < 8; ++r) {
    float inv = 1.0f / lrow[r];
    int row = qb * 64 + wave * 16 + r + 8 * half;
#pragma unroll
    for (int n = 0; n < 4; ++n) {
      int d = n * 16 + lm;
      y[(size_t)(b * SEQ_T + row) * N_EMBD + h * HEAD_D + d] = oacc[n][r] * inv;
    }
  }
}

extern "C" void kernel_launch(void* const* d_in, const int* in_sizes, int n_in,
                              void* d_out, int out_size, void* d_ws,
                              size_t ws_size, hipStream_t stream) {
  const float* x = (const float*)d_in[0];
  const float* W_qkv = (const float*)d_in[1];
  const float* b_qkv = (const float*)d_in[2];
  const float* W_proj = (const float*)d_in[3];
  const float* b_proj = (const float*)d_in[4];
  float* out = (float*)d_out;

  // workspace layout: rope (0.5MB) | qkv (192MB) | y (64MB)
  float* rope = (float*)d_ws;
  float* qkv = rope + (size_t)SEQ_T * 64;
  float* y = qkv + (size_t)BATCH * SEQ_T * THREE_C;

  rope_cache_kernel<<<dim3((SEQ_T * 32 + 255) / 256), dim3(256), 0, stream>>>(
      rope);
  gemm_bias_kernel<<<dim3(THREE_C / 128, BATCH * SEQ_T / 128), dim3(256), 0,
                     stream>>>(x, W_qkv, b_qkv, qkv, BATCH * SEQ_T, THREE_C,
                               N_EMBD);
  flash_attn_kernel<<<dim3(BATCH * N_HEAD, SEQ_T / 64), dim3(128), 0, stream>>>(
      qkv, rope, y);
  gemm_bias_kernel<<<dim3(N_EMBD / 128, BATCH * SEQ_T / 128), dim3(256), 0,
                     stream>>>(y, W_proj, b_proj, out, BATCH * SEQ_T, N_EMBD,
                               N_EMBD);
}